// Attention_88244398063802
// MI455X (gfx1250) — compile-verified
//
#include <hip/hip_runtime.h>
#include <hip/hip_bf16.h>

// ---------------------------------------------------------------------------
// Problem constants (from reference setup_inputs)
// ---------------------------------------------------------------------------
#define B_      32
#define T_ENC   1024
#define T_DEC   128
#define ENC_DIM 512
#define DEC_DIM 256
#define OUT_DIM 512

typedef __attribute__((ext_vector_type(2))) float v2f;
typedef __attribute__((ext_vector_type(8))) float v8f;

// ---------------------------------------------------------------------------
// WMMA f32 16x16x4 fragment layouts (ISA 7.12.2, wave32):
//   A (16x4):  lanes 0-15 hold M=lane, K={k,k+1}; lanes 16-31 hold K={k+2,k+3}
//   B (4x16):  VGPR0 = row k (lanes 0-15) / row k+2 (lanes 16-31),
//              VGPR1 = row k+1 / k+3; column = col0 + (lane&15)
//   C/D 16x16: VGPR r -> row r (lanes 0-15) / row r+8 (lanes 16-31)
// ---------------------------------------------------------------------------

// One K-pass of C[16 x 64] += A[16 x K] * B[K x 64], pointer-induction form.
// Each A fragment is reused across 4 WMMAs (4 accumulators).
__device__ __forceinline__ void gemm_pass_16x64(const float* __restrict__ A, int lda,
                                                const float* __restrict__ Bm, int ldb,
                                                int K, int row0, int col0, int lane,
                                                v8f acc[4]) {
    const float* ap = A + (long)(row0 + (lane & 15)) * lda + ((lane >> 4) << 1);
    const float* bp = Bm + (long)((lane >> 4) << 1) * ldb + col0 + (lane & 15);
    for (int k = 0; k < K; k += 4) {
        v2f a;
        a.x = ap[0];
        a.y = ap[1];
#pragma unroll
        for (int n = 0; n < 4; ++n) {
            v2f b;
            b.x = bp[n * 16];
            b.y = bp[ldb + n * 16];
            acc[n] = __builtin_amdgcn_wmma_f32_16x16x4_f32(
                false, a, false, b, (short)0, acc[n], false, false);
        }
        ap += 4;
        bp += (long)4 * ldb;
    }
}

// ---------------------------------------------------------------------------
// Generic dual-input WMMA GEMM:  C = A*B (+ A2*B2) + bias
// One wave computes one 16x64 output tile (4 f32 accumulators).
// 8 waves / block. M % 16 == 0, N % 64 == 0, K % 4 == 0.
// Tile guard is wave-uniform so EXEC is all-ones at every WMMA.
// ---------------------------------------------------------------------------
__global__ void wmma_gemm2_kernel(const float* __restrict__ A, int lda,
                                  const float* __restrict__ Bm, int ldb, int K,
                                  const float* __restrict__ A2, int lda2,
                                  const float* __restrict__ B2, int ldb2, int K2,
                                  const float* __restrict__ bias,
                                  float* __restrict__ C, int ldc, int M, int N) {
    const int lane = threadIdx.x & 31;
    const int wave = threadIdx.x >> 5;
    const int tile = blockIdx.x * 8 + wave;
    const int tilesN = N >> 6;                 // 16x64 super-tiles along N
    const int totalTiles = (M >> 4) * tilesN;
    if (tile >= totalTiles) return;            // wave-uniform exit

    const int row0 = (tile / tilesN) << 4;
    const int col0 = (tile % tilesN) << 6;

    v8f acc[4] = {{}, {}, {}, {}};
    gemm_pass_16x64(A, lda, Bm, ldb, K, row0, col0, lane, acc);
    if (A2 != nullptr)
        gemm_pass_16x64(A2, lda2, B2, ldb2, K2, row0, col0, lane, acc);

    const int rbase = row0 + ((lane >> 4) << 3);
#pragma unroll
    for (int n = 0; n < 4; ++n) {
        const int colw = col0 + n * 16 + (lane & 15);
        const float cb = (bias != nullptr) ? bias[colw] : 0.0f;
#pragma unroll
        for (int r = 0; r < 8; ++r) {
            C[(long)(rbase + r) * ldc + colw] = acc[n][r] + cb;
        }
    }
}

// ---------------------------------------------------------------------------
// scores[b,t] = sum_d tanh(xW1[b,t,d] + hW2[b,d]) * V[d]
// One wave per (b,t) row; coalesced 256-wide row reads; shfl_xor reduction.
// L2-bandwidth + TRANS-op heavy kernel (33.5 MB + 8.4M tanh per step).
// ---------------------------------------------------------------------------
__global__ void scores_kernel(const float* __restrict__ xW1,
                              const float* __restrict__ hW2,
                              const float* __restrict__ V,
                              float* __restrict__ scores) {
    const int lane = threadIdx.x & 31;
    const int wave = threadIdx.x >> 5;
    const int row  = blockIdx.x * 8 + wave;        // [0, B_*T_ENC)
    if (row >= B_ * T_ENC) return;
    const int b = row >> 10;                        // T_ENC == 1024

    const float* xr = xW1 + (long)row * DEC_DIM;
    const float* hr = hW2 + b * DEC_DIM;

    float s = 0.0f;
#pragma unroll
    for (int j = 0; j < DEC_DIM / 32; ++j) {
        int d = lane + 32 * j;
        s += tanhf(xr[d] + hr[d]) * V[d];
    }
#pragma unroll
    for (int off = 16; off > 0; off >>= 1)
        s += __shfl_xor(s, off, 32);
    if (lane == 0) scores[row] = s;
}

// ---------------------------------------------------------------------------
// In-place softmax over T_ENC per batch row. One 256-thread block per b.
// ---------------------------------------------------------------------------
__global__ void softmax_kernel(float* __restrict__ scores) {
    const int b = blockIdx.x;
    float* row = scores + (long)b * T_ENC;
    const int tid = threadIdx.x, lane = tid & 31, w = tid >> 5;
    __shared__ float red[8];
    __shared__ float bc;

    float m = -3.0e38f;
    for (int i = tid; i < T_ENC; i += 256) m = fmaxf(m, row[i]);
#pragma unroll
    for (int off = 16; off > 0; off >>= 1) m = fmaxf(m, __shfl_xor(m, off, 32));
    if (lane == 0) red[w] = m;
    __syncthreads();
    if (tid == 0) {
        float mm = red[0];
        for (int i = 1; i < 8; ++i) mm = fmaxf(mm, red[i]);
        bc = mm;
    }
    __syncthreads();
    const float mx = bc;
    __syncthreads();

    float s = 0.0f;
    for (int i = tid; i < T_ENC; i += 256) {
        float e = expf(row[i] - mx);
        row[i] = e;
        s += e;
    }
#pragma unroll
    for (int off = 16; off > 0; off >>= 1) s += __shfl_xor(s, off, 32);
    if (lane == 0) red[w] = s;
    __syncthreads();
    if (tid == 0) {
        float ss = 0.0f;
        for (int i = 0; i < 8; ++i) ss += red[i];
        bc = ss;
    }
    __syncthreads();
    const float inv = 1.0f / bc;
    for (int i = tid; i < T_ENC; i += 256) row[i] *= inv;
}

// ---------------------------------------------------------------------------
// Xa[b,e] = sum_t a[b,t] * enc[b,t,e].  One block per b; weights cached in
// LDS; coalesced enc reads (67 MB/step, L2-resident). Clamped prefetch
// (global_prefetch_b8) walks ~64 rows ahead of the reduction.
// ---------------------------------------------------------------------------
__global__ void context_kernel(const float* __restrict__ a,
                               const float* __restrict__ enc,
                               float* __restrict__ Xa) {
    __shared__ float as[T_ENC];
    const int b = blockIdx.x, tid = threadIdx.x;
    for (int i = tid; i < T_ENC; i += 256) as[i] = a[(long)b * T_ENC + i];
    __syncthreads();

    const float* eb = enc + (long)b * T_ENC * ENC_DIM;
    float acc0 = 0.0f, acc1 = 0.0f;
    for (int t = 0; t < T_ENC; t += 8) {
        int pt = t + 64;
        if (pt > T_ENC - 1) pt = T_ENC - 1;   // keep prefetch in-bounds
        __builtin_prefetch(&eb[(long)pt * ENC_DIM + tid], 0, 0);
#pragma unroll
        for (int j = 0; j < 8; ++j) {
            float av = as[t + j];
            acc0 += av * eb[(long)(t + j) * ENC_DIM + tid];
            acc1 += av * eb[(long)(t + j) * ENC_DIM + tid + 256];
        }
    }
    Xa[(long)b * ENC_DIM + tid]       = acc0;
    Xa[(long)b * ENC_DIM + tid + 256] = acc1;
}

// ---------------------------------------------------------------------------
// LSTM cell elementwise + write h into the output sequence.
// ---------------------------------------------------------------------------
__device__ __forceinline__ float hsig(float x) {
    return fminf(fmaxf(0.2f * x + 0.5f, 0.0f), 1.0f);
}

__global__ void lstm_kernel(const float* __restrict__ gates,
                            float* __restrict__ h, float* __restrict__ c,
                            float* __restrict__ out, int t) {
    const int idx = blockIdx.x * 256 + threadIdx.x;   // < B_*OUT_DIM
    if (idx >= B_ * OUT_DIM) return;
    const int b = idx >> 9;                            // OUT_DIM == 512
    const int e = idx & (OUT_DIM - 1);
    const float* g = gates + (long)b * 4 * OUT_DIM;

    float ig = hsig(g[e]);
    float fg = hsig(g[OUT_DIM + e]);
    float gg = tanhf(g[2 * OUT_DIM + e]);
    float og = hsig(g[3 * OUT_DIM + e]);

    float cn = fg * c[idx] + ig * gg;
    float hn = og * tanhf(cn);
    c[idx] = cn;
    h[idx] = hn;
    out[((long)b * T_DEC + t) * OUT_DIM + e] = hn;
}

// ---------------------------------------------------------------------------
// Launcher
// ---------------------------------------------------------------------------
extern "C" void kernel_launch(void* const* d_in, const int* in_sizes, int n_in,
                              void* d_out, int out_size, void* d_ws, size_t ws_size,
                              hipStream_t stream) {
    (void)in_sizes; (void)n_in; (void)out_size; (void)ws_size;

    const float* enc = (const float*)d_in[0];   // [B, T_enc, enc_dim]
    const float* dec = (const float*)d_in[1];   // [B, T_dec, dec_dim]
    const float* W1  = (const float*)d_in[2];   // [enc_dim, dec_dim]
    const float* W2  = (const float*)d_in[3];   // [out_dim, dec_dim]
    const float* b2  = (const float*)d_in[4];   // [dec_dim]
    const float* V   = (const float*)d_in[5];   // [dec_dim]
    const float* W3  = (const float*)d_in[6];   // [dec_dim+out_dim, out_dim]
    const float* b3  = (const float*)d_in[7];   // [out_dim]
    const float* Wx  = (const float*)d_in[8];   // [out_dim, 4*out_dim]
    const float* Uh  = (const float*)d_in[9];   // [out_dim, 4*out_dim]
    const float* bl  = (const float*)d_in[10];  // [4*out_dim]
    const float* h0  = (const float*)d_in[11];  // [B, out_dim]
    const float* c0  = (const float*)d_in[12];  // [B, out_dim]
    float* out = (float*)d_out;                 // [B, T_dec, out_dim]

    // Workspace carve-up (256B-aligned slabs), total ~34.3 MB.
    char* ws = (char*)d_ws;
    size_t off = 0;
    auto carve = [&](size_t bytes) -> float* {
        float* p = (float*)(ws + off);
        off += (bytes + 255) & ~(size_t)255;
        return p;
    };
    float* xW1   = carve((size_t)B_ * T_ENC * DEC_DIM * 4);   // 33.55 MB
    float* hW2   = carve((size_t)B_ * DEC_DIM * 4);
    float* sc    = carve((size_t)B_ * T_ENC * 4);
    float* Xa    = carve((size_t)B_ * ENC_DIM * 4);
    float* z     = carve((size_t)B_ * OUT_DIM * 4);
    float* gates = carve((size_t)B_ * 4 * OUT_DIM * 4);
    float* h     = carve((size_t)B_ * OUT_DIM * 4);
    float* c     = carve((size_t)B_ * OUT_DIM * 4);

    // Initialize recurrent state from inputs (never mutate d_in).
    hipMemcpyAsync(h, h0, (size_t)B_ * OUT_DIM * 4, hipMemcpyDeviceToDevice, stream);
    hipMemcpyAsync(c, c0, (size_t)B_ * OUT_DIM * 4, hipMemcpyDeviceToDevice, stream);

    // One-time: xW1 = enc @ W1   [32768 x 512] * [512 x 256]
    {
        int M = B_ * T_ENC, N = DEC_DIM;
        int blocks = (M / 16) * (N / 64) / 8;   // 1024
        wmma_gemm2_kernel<<<blocks, 256, 0, stream>>>(
            enc, ENC_DIM, W1, DEC_DIM, ENC_DIM,
            nullptr, 0, nullptr, 0, 0,
            nullptr, xW1, DEC_DIM, M, N);
    }

    for (int t = 0; t < T_DEC; ++t) {
        // hW2 = h @ W2 + b2   [32 x 512] * [512 x 256]  -> 8 tiles, 1 block
        wmma_gemm2_kernel<<<1, 256, 0, stream>>>(
            h, OUT_DIM, W2, DEC_DIM, OUT_DIM,
            nullptr, 0, nullptr, 0, 0,
            b2, hW2, DEC_DIM, B_, DEC_DIM);

        // scores = tanh(xW1 + hW2) . V
        scores_kernel<<<(B_ * T_ENC) / 8, 256, 0, stream>>>(xW1, hW2, V, sc);

        // softmax over T_enc (in place)
        softmax_kernel<<<B_, 256, 0, stream>>>(sc);

        // Xa = a^T . enc
        context_kernel<<<B_, 256, 0, stream>>>(sc, enc, Xa);

        // z = concat(x_t, Xa) @ W3 + b3  (dual-input GEMM: no concat needed)
        wmma_gemm2_kernel<<<(B_ / 16) * (OUT_DIM / 64) / 8, 256, 0, stream>>>(
            dec + (long)t * DEC_DIM, T_DEC * DEC_DIM, W3, OUT_DIM, DEC_DIM,
            Xa, ENC_DIM, W3 + (long)DEC_DIM * OUT_DIM, OUT_DIM, ENC_DIM,
            b3, z, OUT_DIM, B_, OUT_DIM);

        // gates = z @ Wx + h @ Uh + b_lstm   [32 x 2048] -> 64 tiles, 8 blocks
        wmma_gemm2_kernel<<<(B_ / 16) * (4 * OUT_DIM / 64) / 8, 256, 0, stream>>>(
            z, OUT_DIM, Wx, 4 * OUT_DIM, OUT_DIM,
            h, OUT_DIM, Uh, 4 * OUT_DIM, OUT_DIM,
            bl, gates, 4 * OUT_DIM, B_, 4 * OUT_DIM);

        // LSTM cell + emit h_t
        lstm_kernel<<<(B_ * OUT_DIM + 255) / 256, 256, 0, stream>>>(gates, h, c, out, t);
    }
}